// Model_13572096655822
// MI455X (gfx1250) — compile-verified
//
#include <hip/hip_runtime.h>

// 8th-order 3D heat stencil, 384^3 fp32, 10 steps, frozen boundaries.
// Bandwidth-bound: ~4.5 GB total HBM traffic -> ~0.2 ms floor @ 23.3 TB/s.
// CDNA5 path: global_load_async_to_lds_b128 (ASYNCcnt) stages halo'd x-y
// tiles into double-buffered LDS; z-direction handled by a per-thread
// register sliding window while the block marches in z.

static constexpr int NG   = 384;
static constexpr int NN   = NG * NG;
static constexpr int NNN  = NG * NG * NG;
static constexpr int RAD  = 4;
static constexpr int TX   = 32;
static constexpr int TY   = 8;
static constexpr int HXT  = TX + 2 * RAD;   // 40 (x extent incl. halo)
static constexpr int HYT  = TY + 2 * RAD;   // 16 (y extent incl. halo)
static constexpr int PLANE = HXT * HYT;     // 640 floats / plane tile
static constexpr int PVEC  = PLANE / 4;     // 160 b128 transfers / plane
static constexpr int NZC   = 8;             // z chunks
static constexpr int ZPC   = (NG - 2 * RAD) / NZC;  // 47 planes per chunk

__device__ __forceinline__ void async_wait_barrier() {
  // Async global->LDS loads are tracked by ASYNCcnt, which the compiler's
  // automatic waitcnt insertion does not cover for the inline-asm path.
  asm volatile("s_wait_asynccnt 0x0" ::: "memory");
  __syncthreads();
}

__global__ __launch_bounds__(TX * TY) void heat_step(
    const float* __restrict__ src, float* __restrict__ dst) {
  __shared__ float smem[2 * PLANE];  // sole LDS object -> LDS offset 0

  // Escape hatch: hand the LDS pointer to an opaque asm so alias analysis
  // must assume the async-load asm (with its "memory" clobber) writes smem.
  // Without this, LLVM proves smem is never stored and folds away all the
  // ds_loads (observed in round 1: ds=0, body deleted).
  asm volatile("" : : "v"(&smem[0]) : "memory");

  const int lx  = threadIdx.x;
  const int ly  = threadIdx.y;
  const int tid = ly * TX + lx;
  const int bx = blockIdx.x, by = blockIdx.y, bz = blockIdx.z;

  const int ox = RAD + bx * TX + lx;
  const int oy = RAD + by * TY + ly;
  const bool valid = (ox < (NG - RAD));  // partial last x-block
  const int z0 = RAD + bz * ZPC;
  const int z1 = z0 + ZPC;

  // Async tile-load geometry: waves 0..4 (tid<160) each move one 16B chunk.
  const int trow = tid / (HXT / 4);                       // 0..15
  const int tcol = tid % (HXT / 4);                       // 0..9
  const int gbase2d = (by * TY + trow) * NG + bx * TX + tcol * 4;
  const unsigned ldsLane = (unsigned)(tid * 16);

  const int cbase = oy * NG + ox;

  auto load_plane = [&](int z, int buf) {
    if (tid < PVEC) {
      int flat = z * NN + gbase2d;
      if (flat > NNN - 4) flat = NNN - 4;  // clamp tail overrun (masked outputs only)
      const float* gp = src + flat;
      unsigned loff = (unsigned)buf * (unsigned)(PLANE * 4) + ldsLane;
      asm volatile("global_load_async_to_lds_b128 %0, %1, off"
                   :: "v"(loff), "v"(gp)
                   : "memory");
    }
  };

  const float C0 = -205.0f / 72.0f;
  const float C1 = 8.0f / 5.0f;
  const float C2 = -1.0f / 5.0f;
  const float C3 = 8.0f / 315.0f;
  const float C4 = -1.0f / 560.0f;
  const float DT = 0.05f / 3.0f;  // CFL / (alpha * (1/hx^2 + 1/hy^2 + 1/hz^2))

  // Prime LDS plane z0 and the per-thread z register window.
  load_plane(z0, 0);
  float zm4 = src[(z0 - 4) * NN + cbase];
  float zm3 = src[(z0 - 3) * NN + cbase];
  float zm2 = src[(z0 - 2) * NN + cbase];
  float zm1 = src[(z0 - 1) * NN + cbase];
  float zp1 = src[(z0 + 1) * NN + cbase];
  float zp2 = src[(z0 + 2) * NN + cbase];
  float zp3 = src[(z0 + 3) * NN + cbase];
  async_wait_barrier();

  const int cidx = (ly + RAD) * HXT + (lx + RAD);
  int cur = 0;

  for (int z = z0; z < z1; ++z) {
    if (z + 1 < z1) load_plane(z + 1, cur ^ 1);      // prefetch next plane
    float zp4 = src[(z + RAD) * NN + cbase];         // coalesced, L2-resident

    const float* sp = smem + cur * PLANE;
    float uc = sp[cidx];
    float s1 = (sp[cidx + 1] + sp[cidx - 1]) +
               (sp[cidx + HXT] + sp[cidx - HXT]) + (zp1 + zm1);
    float s2 = (sp[cidx + 2] + sp[cidx - 2]) +
               (sp[cidx + 2 * HXT] + sp[cidx - 2 * HXT]) + (zp2 + zm2);
    float s3 = (sp[cidx + 3] + sp[cidx - 3]) +
               (sp[cidx + 3 * HXT] + sp[cidx - 3 * HXT]) + (zp3 + zm3);
    float s4 = (sp[cidx + 4] + sp[cidx - 4]) +
               (sp[cidx + 4 * HXT] + sp[cidx - 4 * HXT]) + (zp4 + zm4);
    float lap = 3.0f * C0 * uc + C1 * s1 + C2 * s2 + C3 * s3 + C4 * s4;

    if (valid) dst[z * NN + cbase] = uc + DT * lap;

    zm4 = zm3; zm3 = zm2; zm2 = zm1; zm1 = uc;
    zp1 = zp2; zp2 = zp3; zp3 = zp4;

    async_wait_barrier();  // next plane landed; safe to flip buffers
    cur ^= 1;
  }
}

extern "C" void kernel_launch(void* const* d_in, const int* in_sizes, int n_in,
                              void* d_out, int out_size, void* d_ws, size_t ws_size,
                              hipStream_t stream) {
  const float* u0 = (const float*)d_in[0];
  float* out = (float*)d_out;
  float* ws  = (float*)d_ws;
  const size_t bytes = (size_t)NNN * sizeof(float);

  // Both ping-pong buffers start as u0 so frozen boundaries are correct.
  hipMemcpyAsync(ws,  u0, bytes, hipMemcpyDeviceToDevice, stream);
  hipMemcpyAsync(out, u0, bytes, hipMemcpyDeviceToDevice, stream);

  dim3 grid((NG - 2 * RAD + TX - 1) / TX, (NG - 2 * RAD) / TY, NZC);  // 12 x 47 x 8
  dim3 block(TX, TY, 1);

  const float* s = u0;
  float* d = ws;
  for (int step = 0; step < 10; ++step) {   // u0->ws->out->...; step 10 -> out
    heat_step<<<grid, block, 0, stream>>>(s, d);
    s = d;
    d = (d == ws) ? out : ws;
  }
}